// TTN_FNN_73005854097734
// MI455X (gfx1250) — compile-verified
//
#include <hip/hip_runtime.h>

typedef float v2f __attribute__((ext_vector_type(2)));
typedef float v8f __attribute__((ext_vector_type(8)));

// ---------------------------------------------------------------------------
// Level 0: Fourier features (9-dim) + contraction with W0 (128,9,9,16).
// Per (16-batch-row tile, node): D(16x16) = A(16x84) @ Wflat0(84x16), where
// A[r, i*9+j] = featL[r][i] * featR[r][j], K padded 81 -> 84 (21 steps of 4).
// ---------------------------------------------------------------------------
__global__ __launch_bounds__(256) void ttn_level0(
    const float* __restrict__ x, const float* __restrict__ fminp,
    const float* __restrict__ W0, float* __restrict__ out)
{
    __shared__ float lds[21 * 64];
    const int tid = threadIdx.x;
    const int n   = blockIdx.y;            // output node 0..127

    // Stage W0[n] in WMMA-B fragment order:
    // lds[kk*64 + lane*2 + v] = Wflat0[4*kk + 2*(lane>>4) + v][lane&15]
    for (int t = tid; t < 21 * 64; t += 256) {
        int kk   = t >> 6;
        int pos  = t & 63;
        int ln   = pos >> 1;
        int v    = pos & 1;
        int row  = 4 * kk + 2 * (ln >> 4) + v;   // flattened i*9+j
        int col  = ln & 15;
        lds[t] = (row < 81) ? W0[(n * 81 + row) * 16 + col] : 0.0f;
    }
    __syncthreads();

    const int lane  = tid & 31;
    const int h     = lane >> 4;           // lane half (K offset +2)
    const int r     = lane & 15;           // A-matrix row within tile
    const int bbase = blockIdx.x * 128 + (tid >> 5) * 16;
    const int b     = bbase + r;
    const float fmin = fminp[0];

    const float xl = x[b * 256 + 2 * n]     * fmin;
    const float xr = x[b * 256 + 2 * n + 1] * fmin;
    float fL[9], fR[9];
    fL[0] = 1.0f; fR[0] = 1.0f;
    #pragma unroll
    for (int q = 1; q <= 4; ++q) {
        fL[q]     = __sinf(xl * (float)q);
        fL[4 + q] = __cosf(xl * (float)q);
        fR[q]     = __sinf(xr * (float)q);
        fR[4 + q] = __cosf(xr * (float)q);
    }

    v8f acc = {};
    #pragma unroll
    for (int kk = 0; kk < 21; ++kk) {
        v2f bf = *((const v2f*)(lds + kk * 64 + lane * 2));
        const int k0 = 4 * kk;
        float p0 = (k0     < 81) ? fL[(k0    ) / 9] * fR[(k0    ) % 9] : 0.0f;
        float p1 = (k0 + 1 < 81) ? fL[(k0 + 1) / 9] * fR[(k0 + 1) % 9] : 0.0f;
        float p2 = (k0 + 2 < 81) ? fL[(k0 + 2) / 9] * fR[(k0 + 2) % 9] : 0.0f;
        float p3 = (k0 + 3 < 81) ? fL[(k0 + 3) / 9] * fR[(k0 + 3) % 9] : 0.0f;
        v2f a;
        a.x = h ? p2 : p0;
        a.y = h ? p3 : p1;
        acc = __builtin_amdgcn_wmma_f32_16x16x4_f32(
            false, a, false, bf, (short)0, acc, false, false);
    }

    // D layout: VGPR p -> rows p (lanes 0-15) / p+8 (lanes 16-31), col = lane&15
    #pragma unroll
    for (int p = 0; p < 8; ++p) {
        out[((size_t)(bbase + p + 8 * h) * 128 + n) * 16 + (lane & 15)] = acc[p];
    }
}

// ---------------------------------------------------------------------------
// Generic level (d_in = d_out = 16): per (16-row tile, node)
// D(16x16) = A(16x256) @ Wflat(256x16), A[r, i*16+j] = L[r][i]*R[r][j].
// The lane-half K-offset select on the right operand is hoisted out of the
// K-loop: fRs[2m+v] = fR[4m+v + 2h], so the inner loop is 2 VMUL + 1 DS load
// + 1 WMMA per K-step.
// Input layout (B, 2*n2, 16); output layout (B, n2, 16).
// ---------------------------------------------------------------------------
__global__ __launch_bounds__(256) void ttn_level(
    const float* __restrict__ in, const float* __restrict__ W,
    float* __restrict__ out, int n2)
{
    __shared__ float lds[64 * 64];
    const int tid = threadIdx.x;
    const int n   = blockIdx.y;            // output node 0..n2-1

    // Stage Wflat[n] (= W[n] viewed as (256,16) row-major) in fragment order.
    for (int t = tid; t < 64 * 64; t += 256) {
        int kk  = t >> 6;
        int pos = t & 63;
        int ln  = pos >> 1;
        int v   = pos & 1;
        int row = 4 * kk + 2 * (ln >> 4) + v;    // 0..255  (= i*16+j)
        int col = ln & 15;
        lds[t] = W[((size_t)n * 256 + row) * 16 + col];
    }
    __syncthreads();

    const int lane  = tid & 31;
    const int h     = lane >> 4;
    const int r     = lane & 15;
    const int bbase = blockIdx.x * 128 + (tid >> 5) * 16;
    const int b     = bbase + r;

    const float* Lp = in + ((size_t)b * 2 * n2 + 2 * n) * 16;
    float fL[16], fR[16];
    #pragma unroll
    for (int e = 0; e < 16; e += 4) {
        float4 lv = *((const float4*)(Lp + e));
        fL[e] = lv.x; fL[e + 1] = lv.y; fL[e + 2] = lv.z; fL[e + 3] = lv.w;
        float4 rv = *((const float4*)(Lp + 16 + e));
        fR[e] = rv.x; fR[e + 1] = rv.y; fR[e + 2] = rv.z; fR[e + 3] = rv.w;
    }

    // Hoisted lane-half select: fRs[2m+v] = fR[4m + v + 2h], m=0..3, v=0..1.
    float fRs[8];
    #pragma unroll
    for (int m = 0; m < 4; ++m) {
        fRs[2 * m]     = h ? fR[4 * m + 2] : fR[4 * m    ];
        fRs[2 * m + 1] = h ? fR[4 * m + 3] : fR[4 * m + 1];
    }

    v8f acc = {};
    #pragma unroll
    for (int kk = 0; kk < 64; ++kk) {
        v2f bf = *((const v2f*)(lds + kk * 64 + lane * 2));
        const int i = kk >> 2;
        const int m = kk & 3;
        v2f a;
        a.x = fL[i] * fRs[2 * m];
        a.y = fL[i] * fRs[2 * m + 1];
        acc = __builtin_amdgcn_wmma_f32_16x16x4_f32(
            false, a, false, bf, (short)0, acc, false, false);
    }

    #pragma unroll
    for (int p = 0; p < 8; ++p) {
        out[((size_t)(bbase + p + 8 * h) * n2 + n) * 16 + (lane & 15)] = acc[p];
    }
}

// ---------------------------------------------------------------------------
// Level 7: (B,2,16) x W7(1,16,16,1) -> (B,) — tiny, scalar per thread.
// ---------------------------------------------------------------------------
__global__ __launch_bounds__(256) void ttn_level7(
    const float* __restrict__ in, const float* __restrict__ W7,
    float* __restrict__ out)
{
    const int b = blockIdx.x * 256 + threadIdx.x;
    const float* Lp = in + (size_t)b * 32;
    float acc = 0.0f;
    #pragma unroll
    for (int i = 0; i < 16; ++i) {
        float t = 0.0f;
        #pragma unroll
        for (int j = 0; j < 16; ++j)
            t = fmaf(W7[i * 16 + j], Lp[16 + j], t);
        acc = fmaf(Lp[i], t, acc);
    }
    out[b] = acc;
}

extern "C" void kernel_launch(void* const* d_in, const int* in_sizes, int n_in,
                              void* d_out, int out_size, void* d_ws, size_t ws_size,
                              hipStream_t stream)
{
    (void)in_sizes; (void)n_in; (void)out_size; (void)ws_size;
    const int B = 16384;

    const float* x    = (const float*)d_in[0];
    const float* fmin = (const float*)d_in[1];
    const float* W0   = (const float*)d_in[2];
    const float* W1   = (const float*)d_in[3];
    const float* W2   = (const float*)d_in[4];
    const float* W3   = (const float*)d_in[5];
    const float* W4   = (const float*)d_in[6];
    const float* W5   = (const float*)d_in[7];
    const float* W6   = (const float*)d_in[8];
    const float* W7   = (const float*)d_in[9];

    float* r0  = (float*)d_ws;                                   // 128 MiB region
    float* r1  = (float*)((char*)d_ws + (size_t)134217728);      //  64 MiB region
    float* out = (float*)d_out;

    ttn_level0<<<dim3(B / 128, 128), 256, 0, stream>>>(x, fmin, W0, r0);
    ttn_level <<<dim3(B / 128,  64), 256, 0, stream>>>(r0, W1, r1, 64);
    ttn_level <<<dim3(B / 128,  32), 256, 0, stream>>>(r1, W2, r0, 32);
    ttn_level <<<dim3(B / 128,  16), 256, 0, stream>>>(r0, W3, r1, 16);
    ttn_level <<<dim3(B / 128,   8), 256, 0, stream>>>(r1, W4, r0,  8);
    ttn_level <<<dim3(B / 128,   4), 256, 0, stream>>>(r0, W5, r1,  4);
    ttn_level <<<dim3(B / 128,   2), 256, 0, stream>>>(r1, W6, r0,  2);
    ttn_level7<<<B / 256, 256, 0, stream>>>(r0, W7, out);
}